// MoELayer_34291018891540
// MI455X (gfx1250) — compile-verified
//
#include <hip/hip_runtime.h>
#include <hip/hip_bf16.h>
#include <math.h>

// ---------------- problem constants ----------------
constexpr int NE   = 8;        // experts
constexpr int HID  = 768;      // hidden
constexpr int FFN  = 3072;     // ffn
constexpr int T    = 16384;    // tokens (B*S = 4*4096)
constexpr int TEXP = T;        // per-expert list capacity (worst case)

constexpr int TILE_M   = 16;   // tokens per expert tile
constexpr int CHUNK    = 768;  // FFN chunk held in LDS
constexpr int NCHUNKS  = FFN / CHUNK;          // 4
constexpr int NT_WAVE  = (CHUNK / 16) / 8;     // 6 N-tiles per wave (8 waves)

typedef float v2f __attribute__((ext_vector_type(2)));
typedef float v8f __attribute__((ext_vector_type(8)));

// Branch-free tanh-GELU: tanh(u) = 1 - 2/(exp(2u)+1).
// __expf -> single v_exp_f32; saturates correctly at +/-inf.
__device__ __forceinline__ float gelu_tanh(float v) {
    const float k0 = 0.7978845608028654f;   // sqrt(2/pi)
    const float k1 = 0.044715f;
    const float u  = k0 * (v + k1 * v * v * v);
    const float th = 1.0f - 2.0f / (__expf(2.0f * u) + 1.0f);
    return 0.5f * v * (1.0f + th);
}

// ---------------- kernel 0: init workspace counts ----------------
__global__ void moe_init_kernel(int* __restrict__ cnt) {
    if (threadIdx.x < NE) cnt[threadIdx.x] = 0;
}

// ---------------- kernel 1: router (one wave32 per token) ----------------
__global__ __launch_bounds__(256)
void moe_router_kernel(const float* __restrict__ x,      // [T,HID]
                       const float* __restrict__ rw,     // [HID,NE]
                       const float* __restrict__ rb,     // [NE]
                       float* __restrict__ o_rweights,   // [T,NE]
                       float* __restrict__ o_dmask,      // [T,NE]
                       float* __restrict__ o_logits,     // [T,NE]
                       float* __restrict__ comb,         // [T,HID] (zeroed here)
                       int*   __restrict__ cnt,          // [NE]
                       int*   __restrict__ toklist,      // [NE,TEXP]
                       float* __restrict__ gatelist)     // [NE,TEXP]
{
    const int lane = threadIdx.x & 31;
    const int t = blockIdx.x * (256 / 32) + (threadIdx.x >> 5);
    if (t >= T) return;

    float p[NE];
#pragma unroll
    for (int e = 0; e < NE; ++e) p[e] = 0.f;

    // vectorized GEMV: each lane processes float4 chunks of the token row
    const float4* xr4 = (const float4*)(x + (size_t)t * HID);
    for (int h4 = lane; h4 < HID / 4; h4 += 32) {
        const float4 xv = xr4[h4];
        const float* wr = rw + (size_t)h4 * 4 * NE;
#pragma unroll
        for (int e = 0; e < NE; ++e) p[e] += xv.x * wr[0 * NE + e];
#pragma unroll
        for (int e = 0; e < NE; ++e) p[e] += xv.y * wr[1 * NE + e];
#pragma unroll
        for (int e = 0; e < NE; ++e) p[e] += xv.z * wr[2 * NE + e];
#pragma unroll
        for (int e = 0; e < NE; ++e) p[e] += xv.w * wr[3 * NE + e];
    }
#pragma unroll
    for (int e = 0; e < NE; ++e) {
#pragma unroll
        for (int off = 16; off > 0; off >>= 1)
            p[e] += __shfl_xor(p[e], off, 32);
        p[e] += rb[e];
    }

    // zero the combined row with b128 stores (expert kernel scatter-adds into it)
    {
        float4* cr4 = (float4*)(comb + (size_t)t * HID);
        const float4 z = make_float4(0.f, 0.f, 0.f, 0.f);
        for (int h4 = lane; h4 < HID / 4; h4 += 32) cr4[h4] = z;
    }

    if (lane == 0) {
        // logits
#pragma unroll
        for (int e = 0; e < NE; ++e) o_logits[(size_t)t * NE + e] = p[e];
        // full softmax -> routing_weights
        float mx = p[0];
#pragma unroll
        for (int e = 1; e < NE; ++e) mx = fmaxf(mx, p[e]);
        float pr[NE]; float s = 0.f;
#pragma unroll
        for (int e = 0; e < NE; ++e) { pr[e] = __expf(p[e] - mx); s += pr[e]; }
        const float inv = 1.f / s;
#pragma unroll
        for (int e = 0; e < NE; ++e) o_rweights[(size_t)t * NE + e] = pr[e] * inv;
        // top-2 (first occurrence wins ties, matching lax.top_k)
        int i0 = 0; float v0 = p[0];
#pragma unroll
        for (int e = 1; e < NE; ++e) if (p[e] > v0) { v0 = p[e]; i0 = e; }
        int i1 = -1; float v1 = -3.4e38f;
#pragma unroll
        for (int e = 0; e < NE; ++e) if (e != i0 && p[e] > v1) { v1 = p[e]; i1 = e; }
        const float m2 = fmaxf(v0, v1);
        const float e0 = __expf(v0 - m2), e1 = __expf(v1 - m2);
        const float g0 = e0 / (e0 + e1), g1 = e1 / (e0 + e1);
        float dm[NE];
#pragma unroll
        for (int e = 0; e < NE; ++e) dm[e] = 0.f;
        dm[i0] = g0; dm[i1] = g1;
#pragma unroll
        for (int e = 0; e < NE; ++e) o_dmask[(size_t)t * NE + e] = dm[e];
        // append to expert lists
        int pos0 = atomicAdd(&cnt[i0], 1);
        toklist[i0 * TEXP + pos0] = t; gatelist[i0 * TEXP + pos0] = g0;
        int pos1 = atomicAdd(&cnt[i1], 1);
        toklist[i1 * TEXP + pos1] = t; gatelist[i1 * TEXP + pos1] = g1;
    }
}

// ---------------- kernel 2: fused expert FFN via f32 WMMA ----------------
__global__ __launch_bounds__(256)
void moe_expert_kernel(const float* __restrict__ x,     // [T,HID]
                       const float* __restrict__ w1,    // [NE,HID,FFN]
                       const float* __restrict__ b1,    // [NE,FFN]
                       const float* __restrict__ w2,    // [NE,FFN,HID]
                       const float* __restrict__ b2,    // [NE,HID]
                       const int*   __restrict__ cnt,
                       const int*   __restrict__ toklist,
                       const float* __restrict__ gatelist,
                       float* __restrict__ comb)        // [T,HID]
{
    __shared__ float x_lds[TILE_M * HID];     // 48 KB
    __shared__ float h_lds[TILE_M * CHUNK];   // 48 KB
    __shared__ int   s_tok[TILE_M];
    __shared__ float s_gate[TILE_M];

    const int e    = blockIdx.y;
    const int tile = blockIdx.x;
    const int n_e  = cnt[e];
    if (tile * TILE_M >= n_e) return;

    const int tid  = threadIdx.x;
    const int wave = tid >> 5;
    const int lane = tid & 31;
    const int nsub = lane & 15;             // N index (B/C/D), M index (A)
    const int kb   = (lane >> 4) << 1;      // K pair: 0 or 2
    const int mhi  = (lane >> 4) << 3;      // +8 row offset for C/D VGPR rows

    if (tid < TILE_M) {
        const int idx = tile * TILE_M + tid;
        if (idx < n_e) {
            s_tok[tid]  = toklist[e * TEXP + idx];
            s_gate[tid] = gatelist[e * TEXP + idx];
        } else { s_tok[tid] = -1; s_gate[tid] = 0.f; }
    }
    __syncthreads();

    // stage X tile with b128 loads/stores (pad missing tokens with zeros)
    {
        constexpr int ROW4 = HID / 4;               // 192 float4 per row
        constexpr int ELEMS4 = TILE_M * ROW4;       // 3072
        float4* xl4 = (float4*)x_lds;
        for (int idx = tid; idx < ELEMS4; idx += 256) {
            const int m  = idx / ROW4;
            const int h4 = idx - m * ROW4;
            const int tk = s_tok[m];
            float4 vv = make_float4(0.f, 0.f, 0.f, 0.f);
            if (tk >= 0)
                vv = ((const float4*)(x + (size_t)tk * HID))[h4];
            xl4[idx] = vv;
        }
    }
    __syncthreads();

    const float* w1e = w1 + (size_t)e * HID * FFN;
    const float* w2e = w2 + (size_t)e * FFN * HID;
    const float* b1e = b1 + (size_t)e * FFN;
    const float* b2e = b2 + (size_t)e * HID;

    // GEMM2 accumulators (live across FFN chunks): Y columns wave*6+j
    v8f yacc[NT_WAVE];
#pragma unroll
    for (int j = 0; j < NT_WAVE; ++j) {
        const float bv = b2e[(wave * NT_WAVE + j) * 16 + nsub];
        yacc[j] = (v8f){bv, bv, bv, bv, bv, bv, bv, bv};
    }

    for (int c = 0; c < NCHUNKS; ++c) {
        // ---- GEMM1: H_chunk = gelu(X @ w1[:, c*768 : (c+1)*768] + b1) ----
        v8f hacc[NT_WAVE];
#pragma unroll
        for (int j = 0; j < NT_WAVE; ++j) {
            const float bv = b1e[c * CHUNK + (wave * NT_WAVE + j) * 16 + nsub];
            hacc[j] = (v8f){bv, bv, bv, bv, bv, bv, bv, bv};
        }
        for (int k = 0; k < HID; k += 4) {
            const v2f a = *(const v2f*)&x_lds[nsub * HID + k + kb];
#pragma unroll
            for (int j = 0; j < NT_WAVE; ++j) {
                const int col = c * CHUNK + (wave * NT_WAVE + j) * 16 + nsub;
                v2f b;
                b.x = w1e[(size_t)(k + kb) * FFN + col];
                b.y = w1e[(size_t)(k + kb + 1) * FFN + col];
                hacc[j] = __builtin_amdgcn_wmma_f32_16x16x4_f32(
                    false, a, false, b, (short)0, hacc[j], false, false);
            }
        }
        // gelu + store chunk into LDS (previous chunk's consumers finished at
        // the barrier at the end of the prior iteration)
#pragma unroll
        for (int j = 0; j < NT_WAVE; ++j) {
            const int nl = (wave * NT_WAVE + j) * 16 + nsub;
#pragma unroll
            for (int i = 0; i < 8; ++i) {
                const int m = i + mhi;
                h_lds[m * CHUNK + nl] = gelu_tanh(hacc[j][i]);
            }
        }
        __syncthreads();

        // ---- GEMM2 partial: Y += H_chunk @ w2[c*768 : (c+1)*768, :] ----
        for (int k = 0; k < CHUNK; k += 4) {
            const v2f a = *(const v2f*)&h_lds[nsub * CHUNK + k + kb];
            const size_t f0 = (size_t)(c * CHUNK + k + kb);
#pragma unroll
            for (int j = 0; j < NT_WAVE; ++j) {
                const int col = (wave * NT_WAVE + j) * 16 + nsub;
                v2f b;
                b.x = w2e[f0 * HID + col];
                b.y = w2e[(f0 + 1) * HID + col];
                yacc[j] = __builtin_amdgcn_wmma_f32_16x16x4_f32(
                    false, a, false, b, (short)0, yacc[j], false, false);
            }
        }
        __syncthreads();
    }

    // gate-scaled scatter-add into combined (each token receives exactly 2
    // contributions overall -> commutative 2-term fp sum, deterministic)
#pragma unroll
    for (int j = 0; j < NT_WAVE; ++j) {
        const int col = (wave * NT_WAVE + j) * 16 + nsub;
#pragma unroll
        for (int i = 0; i < 8; ++i) {
            const int m = i + mhi;
            const int tk = s_tok[m];
            if (tk >= 0)
                atomicAdd(&comb[(size_t)tk * HID + col], s_gate[m] * yacc[j][i]);
        }
    }
}

// ---------------- kernel 3: aux loss (deterministic tree reduction) ----------------
__global__ __launch_bounds__(256)
void moe_aux_kernel(const float* __restrict__ rwts,    // [T,NE]
                    const float* __restrict__ dmask,   // [T,NE]
                    float* __restrict__ o_aux)
{
    __shared__ float red[256];
    const int tid = threadIdx.x;
    float sa[NE], sb[NE];
#pragma unroll
    for (int e = 0; e < NE; ++e) { sa[e] = 0.f; sb[e] = 0.f; }
    for (int t = tid; t < T; t += 256) {
#pragma unroll
        for (int e = 0; e < NE; ++e) {
            sa[e] += rwts[(size_t)t * NE + e];
            sb[e] += dmask[(size_t)t * NE + e];
        }
    }
    float aux = 0.f;
    for (int e = 0; e < NE; ++e) {
        red[tid] = sa[e]; __syncthreads();
        for (int s = 128; s > 0; s >>= 1) {
            if (tid < s) red[tid] += red[tid + s];
            __syncthreads();
        }
        const float ta = red[0]; __syncthreads();
        red[tid] = sb[e]; __syncthreads();
        for (int s = 128; s > 0; s >>= 1) {
            if (tid < s) red[tid] += red[tid + s];
            __syncthreads();
        }
        const float tb = red[0]; __syncthreads();
        if (tid == 0) aux += (ta / (float)T) * (tb / (float)T);
    }
    if (tid == 0) o_aux[0] = aux * (float)NE;
}

// ---------------- host launch ----------------
extern "C" void kernel_launch(void* const* d_in, const int* in_sizes, int n_in,
                              void* d_out, int out_size, void* d_ws, size_t ws_size,
                              hipStream_t stream) {
    const float* x  = (const float*)d_in[0];   // hidden_states [B,S,HID]
    const float* rw = (const float*)d_in[1];   // router_w [HID,NE]
    const float* rb = (const float*)d_in[2];   // router_b [NE]
    const float* w1 = (const float*)d_in[3];   // [NE,HID,FFN]
    const float* b1 = (const float*)d_in[4];   // [NE,FFN]
    const float* w2 = (const float*)d_in[5];   // [NE,FFN,HID]
    const float* b2 = (const float*)d_in[6];   // [NE,HID]

    float* out     = (float*)d_out;
    float* o_comb  = out;                              // [T,HID]
    float* o_rwts  = o_comb + (size_t)T * HID;         // [T,NE]
    float* o_dmask = o_rwts + (size_t)T * NE;          // [T,NE]
    float* o_aux   = o_dmask + (size_t)T * NE;         // scalar
    float* o_logit = o_aux + 1;                        // [T,NE]

    int*   cnt      = (int*)d_ws;                      // [16] (8 used)
    int*   toklist  = cnt + 16;                        // [NE*TEXP]
    float* gatelist = (float*)(toklist + NE * TEXP);   // [NE*TEXP]

    moe_init_kernel<<<1, 32, 0, stream>>>(cnt);

    moe_router_kernel<<<T / 8, 256, 0, stream>>>(
        x, rw, rb, o_rwts, o_dmask, o_logit, o_comb, cnt, toklist, gatelist);

    dim3 g2(T / TILE_M, NE);   // 1024 tiles x 8 experts; excess tiles early-exit
    moe_expert_kernel<<<g2, 256, 0, stream>>>(
        x, w1, b1, w2, b2, cnt, toklist, gatelist, o_comb);

    moe_aux_kernel<<<1, 256, 0, stream>>>(o_rwts, o_dmask, o_aux);
}